// TopKSAE_27066883899542
// MI455X (gfx1250) — compile-verified
//
#include <hip/hip_runtime.h>

#define B_ROWS 4096
#define D_IN   2048
#define D_H    32768

typedef float v2f __attribute__((ext_vector_type(2)));
typedef float v8f __attribute__((ext_vector_type(8)));

#define LDS_STRIDE 36              // 32 K-floats + 4 pad: conflict-free b64 reads
#define ABUF (64 * LDS_STRIDE)     // one A buffer: 64 rows
#define BBUF (128 * LDS_STRIDE)    // one B buffer: 128 rows

// ---------------------------------------------------------------------------
// NT GEMM: C[M,N] = (A[M,K] - center[K]?) * B[N,K]^T + bias[N]
// Block: 256 threads (8 waves, 4(M) x 2(N)), tile 64(M) x 128(N).
// Each wave: 16x64 of C = 4 independent v8f accumulators (4 WMMA chains).
// K staged 32 at a time with double-buffered LDS (1 barrier / K-step).
// ---------------------------------------------------------------------------
template <bool CENTER>
__global__ __launch_bounds__(256) void gemm_nt(
    const float* __restrict__ A, const float* __restrict__ Bm,
    const float* __restrict__ bias, const float* __restrict__ center,
    float* __restrict__ C, int M, int N, int K) {
  __shared__ float Als[2 * ABUF];
  __shared__ float Bls[2 * BBUF];

  const int tid = threadIdx.x;
  const int bn = blockIdx.x * 128;
  const int bm = blockIdx.y * 64;

  const int lrow = tid >> 3;         // 0..31
  const int lcol4 = (tid & 7) << 2;  // 0..28 step 4

  const int lane = tid & 31;
  const int wave = tid >> 5;         // 0..7
  const int wm = wave >> 1;          // 0..3  (16-row sub-tile)
  const int wn = wave & 1;           // 0..1  (64-col sub-tile)
  const int l15 = lane & 15;
  const int h2 = (lane >> 4) << 1;   // A/B frag: K=0,1 (lanes 0-15) vs K=2,3

  const float* Ag0 = A + (size_t)(bm + lrow) * K + lcol4;
  const float* Ag1 = Ag0 + (size_t)32 * K;
  const float* Bg = Bm + (size_t)(bn + lrow) * K + lcol4;

  float* AlsSt = &Als[lrow * LDS_STRIDE + lcol4];
  float* BlsSt = &Bls[lrow * LDS_STRIDE + lcol4];

  const float* Ald = &Als[(wm * 16 + l15) * LDS_STRIDE + h2];
  const float* Bld = &Bls[(wn * 64 + l15) * LDS_STRIDE + h2];

  v8f acc0 = {}, acc1 = {}, acc2 = {}, acc3 = {};
  float4 ar0, ar1, br0, br1, br2, br3;

  // ---- stage tile 0: global -> regs -> LDS buf0
  {
    ar0 = *(const float4*)(Ag0);
    ar1 = *(const float4*)(Ag1);
    if (CENTER) {
      float4 cv = *(const float4*)(center + lcol4);
      ar0.x -= cv.x; ar0.y -= cv.y; ar0.z -= cv.z; ar0.w -= cv.w;
      ar1.x -= cv.x; ar1.y -= cv.y; ar1.z -= cv.z; ar1.w -= cv.w;
    }
    br0 = *(const float4*)(Bg);
    br1 = *(const float4*)(Bg + (size_t)32 * K);
    br2 = *(const float4*)(Bg + (size_t)64 * K);
    br3 = *(const float4*)(Bg + (size_t)96 * K);
    *(float4*)(AlsSt) = ar0;
    *(float4*)(AlsSt + 32 * LDS_STRIDE) = ar1;
    *(float4*)(BlsSt) = br0;
    *(float4*)(BlsSt + 32 * LDS_STRIDE) = br1;
    *(float4*)(BlsSt + 64 * LDS_STRIDE) = br2;
    *(float4*)(BlsSt + 96 * LDS_STRIDE) = br3;
  }
  // ---- issue global loads for tile 1 before first barrier
  if (K > 32) {
    ar0 = *(const float4*)(Ag0 + 32);
    ar1 = *(const float4*)(Ag1 + 32);
    if (CENTER) {
      float4 cv = *(const float4*)(center + 32 + lcol4);
      ar0.x -= cv.x; ar0.y -= cv.y; ar0.z -= cv.z; ar0.w -= cv.w;
      ar1.x -= cv.x; ar1.y -= cv.y; ar1.z -= cv.z; ar1.w -= cv.w;
    }
    br0 = *(const float4*)(Bg + 32);
    br1 = *(const float4*)(Bg + 32 + (size_t)32 * K);
    br2 = *(const float4*)(Bg + 32 + (size_t)64 * K);
    br3 = *(const float4*)(Bg + 32 + (size_t)96 * K);
  }
  __syncthreads();

  int cur = 0;
  for (int k0 = 0; k0 < K; k0 += 32) {
    // ---- compute on buf[cur]
    const float* Ap = Ald + cur * ABUF;
    const float* Bp = Bld + cur * BBUF;
#pragma unroll
    for (int kk = 0; kk < 32; kk += 4) {
      v2f a  = *(const v2f*)(Ap + kk);
      v2f b0 = *(const v2f*)(Bp + kk);
      v2f b1 = *(const v2f*)(Bp + kk + 16 * LDS_STRIDE);
      v2f b2 = *(const v2f*)(Bp + kk + 32 * LDS_STRIDE);
      v2f b3 = *(const v2f*)(Bp + kk + 48 * LDS_STRIDE);
      acc0 = __builtin_amdgcn_wmma_f32_16x16x4_f32(false, a, false, b0,
                                                   (short)0, acc0, false, false);
      acc1 = __builtin_amdgcn_wmma_f32_16x16x4_f32(false, a, false, b1,
                                                   (short)0, acc1, false, false);
      acc2 = __builtin_amdgcn_wmma_f32_16x16x4_f32(false, a, false, b2,
                                                   (short)0, acc2, false, false);
      acc3 = __builtin_amdgcn_wmma_f32_16x16x4_f32(false, a, false, b3,
                                                   (short)0, acc3, false, false);
    }
    // ---- stage tile (k0+32) regs -> the other LDS buffer (1-cur); issue loads for (k0+64)
    if (k0 + 32 < K) {
      float* As = AlsSt + (1 - cur) * ABUF;
      float* Bs = BlsSt + (1 - cur) * BBUF;
      *(float4*)(As) = ar0;
      *(float4*)(As + 32 * LDS_STRIDE) = ar1;
      *(float4*)(Bs) = br0;
      *(float4*)(Bs + 32 * LDS_STRIDE) = br1;
      *(float4*)(Bs + 64 * LDS_STRIDE) = br2;
      *(float4*)(Bs + 96 * LDS_STRIDE) = br3;
      const int kn = k0 + 64;
      if (kn < K) {
        ar0 = *(const float4*)(Ag0 + kn);
        ar1 = *(const float4*)(Ag1 + kn);
        if (CENTER) {
          float4 cv = *(const float4*)(center + kn + lcol4);
          ar0.x -= cv.x; ar0.y -= cv.y; ar0.z -= cv.z; ar0.w -= cv.w;
          ar1.x -= cv.x; ar1.y -= cv.y; ar1.z -= cv.z; ar1.w -= cv.w;
        }
        br0 = *(const float4*)(Bg + kn);
        br1 = *(const float4*)(Bg + kn + (size_t)32 * K);
        br2 = *(const float4*)(Bg + kn + (size_t)64 * K);
        br3 = *(const float4*)(Bg + kn + (size_t)96 * K);
        if (kn + 32 < K) {  // nudge tile k0+96 toward L2
          __builtin_prefetch(Ag0 + kn + 32, 0, 1);
          __builtin_prefetch(Bg + kn + 32, 0, 1);
        }
      }
    }
    __syncthreads();
    cur ^= 1;
  }

  // ---- epilogue: bias + store (C/D layout: VGPR r -> rows r, r+8)
  const int mg = bm + wm * 16 + ((lane >> 4) << 3);
  v8f accs[4] = {acc0, acc1, acc2, acc3};
#pragma unroll
  for (int j = 0; j < 4; ++j) {
    const int ng = bn + wn * 64 + j * 16 + l15;
    const float bvv = bias[ng];
#pragma unroll
    for (int r = 0; r < 8; ++r)
      C[(size_t)(mg + r) * N + ng] = accs[j][r] + bvv;
  }
}

// ---------------------------------------------------------------------------
// Per-row top-k(64) + relu, in place. Keys staged in 128KB dynamic LDS.
// ---------------------------------------------------------------------------
__device__ __forceinline__ unsigned f2k(float f) {
  unsigned u = __float_as_uint(f);
  return (u & 0x80000000u) ? ~u : (u | 0x80000000u);  // monotone float order
}

__global__ __launch_bounds__(256) void topk_relu(
    float* __restrict__ z, unsigned* __restrict__ l0_count,
    const int* __restrict__ kptr) {
  extern __shared__ unsigned skey[];  // D_H keys (128 KB)
  __shared__ unsigned red[256];
  __shared__ unsigned tiectr;

  const int tid = threadIdx.x;
  const int K = kptr[0];
  float* row = z + (size_t)blockIdx.x * D_H;

  for (int i = tid; i < D_H; i += 256) skey[i] = f2k(row[i]);
  __syncthreads();

  // bisection: largest T with |{key >= T}| >= K   (uniform across block)
  unsigned lo = 0u, hi = 0xFFFFFFFFu;
  for (int it = 0; it < 33; ++it) {
    if (lo >= hi) break;
    unsigned mid = lo + ((hi - lo) >> 1) + 1u;
    unsigned c = 0;
    for (int i = tid; i < D_H; i += 256) c += (skey[i] >= mid) ? 1u : 0u;
    red[tid] = c;
    __syncthreads();
    for (int s = 128; s > 0; s >>= 1) {
      if (tid < s) red[tid] += red[tid + s];
      __syncthreads();
    }
    unsigned cnt = red[0];
    __syncthreads();
    if (cnt >= (unsigned)K) lo = mid; else hi = mid - 1u;
  }
  const unsigned T = lo;

  // strictly-greater count (sizes the tie quota)
  unsigned c = 0;
  for (int i = tid; i < D_H; i += 256) c += (skey[i] > T) ? 1u : 0u;
  red[tid] = c;
  __syncthreads();
  for (int s = 128; s > 0; s >>= 1) {
    if (tid < s) red[tid] += red[tid + s];
    __syncthreads();
  }
  const unsigned c_gt = red[0];
  if (tid == 0) tiectr = 0u;
  __syncthreads();
  const unsigned need_eq = (unsigned)K - c_gt;

  unsigned myl0 = 0;
  for (int i = tid; i < D_H; i += 256) {
    unsigned key = skey[i];
    float val = row[i];
    bool sel = key > T;
    if (!sel && key == T) sel = (atomicAdd(&tiectr, 1u) < need_eq);
    bool pos = sel && (val > 0.0f);
    myl0 += pos ? 1u : 0u;
    row[i] = pos ? val : 0.0f;  // relu(topk), zero elsewhere
  }
  red[tid] = myl0;
  __syncthreads();
  for (int s = 128; s > 0; s >>= 1) {
    if (tid < s) red[tid] += red[tid + s];
    __syncthreads();
  }
  if (tid == 0) atomicAdd(l0_count, red[0]);
}

// ---------------------------------------------------------------------------
// mean((xhat - x)^2) partial reduction
// ---------------------------------------------------------------------------
__global__ __launch_bounds__(256) void recon_loss(
    const float* __restrict__ xhat, const float* __restrict__ x,
    float* __restrict__ acc) {
  __shared__ float red[256];
  const int tid = threadIdx.x;
  const size_t base = (size_t)blockIdx.x * 2048;
  float s = 0.f;
  for (int i = tid; i < 2048; i += 256) {
    float d = xhat[base + i] - x[base + i];
    s += d * d;
  }
  red[tid] = s;
  __syncthreads();
  for (int st = 128; st > 0; st >>= 1) {
    if (tid < st) red[tid] += red[tid + st];
    __syncthreads();
  }
  if (tid == 0) atomicAdd(acc, red[0]);
}

__global__ void init_accum(float* lossacc, unsigned* l0cnt) {
  *lossacc = 0.f;
  *l0cnt = 0u;
}

__global__ void finalize_k(const float* lossacc, const unsigned* l0cnt,
                           float* lossout, float* l0out) {
  *lossout = *lossacc / ((float)B_ROWS * (float)D_IN);
  *l0out = (float)(*l0cnt) / (float)B_ROWS;
}

// ---------------------------------------------------------------------------
extern "C" void kernel_launch(void* const* d_in, const int* in_sizes, int n_in,
                              void* d_out, int out_size, void* d_ws,
                              size_t ws_size, hipStream_t stream) {
  const float* x = (const float*)d_in[0];      // [4096,2048]
  const float* W_enc = (const float*)d_in[1];  // [32768,2048]
  const float* b_enc = (const float*)d_in[2];  // [32768]
  const float* W_dec = (const float*)d_in[3];  // [2048,32768]
  const float* b_dec = (const float*)d_in[4];  // [2048]
  const int* kptr = (const int*)d_in[5];       // k = 64

  float* out = (float*)d_out;
  float* xhat = out;                           // [4096,2048]
  float* z = out + (size_t)B_ROWS * D_IN;      // [4096,32768]
  float* loss = z + (size_t)B_ROWS * D_H;      // scalar
  float* l0 = loss + 1;                        // scalar

  float* lossacc = (float*)d_ws;
  unsigned* l0cnt = (unsigned*)((char*)d_ws + sizeof(float));

  init_accum<<<1, 1, 0, stream>>>(lossacc, l0cnt);

  // encoder: pre_acts (into z region) = (x - b_dec) @ W_enc^T + b_enc
  gemm_nt<true><<<dim3(D_H / 128, B_ROWS / 64), 256, 0, stream>>>(
      x, W_enc, b_enc, b_dec, z, B_ROWS, D_H, D_IN);

  // per-row top-k + relu in place; l0 count
  topk_relu<<<B_ROWS, 256, D_H * sizeof(unsigned), stream>>>(z, l0cnt, kptr);

  // decoder: xhat = z @ W_dec^T + b_dec
  gemm_nt<false><<<dim3(D_IN / 128, B_ROWS / 64), 256, 0, stream>>>(
      z, W_dec, b_dec, nullptr, xhat, B_ROWS, D_H == D_H ? D_IN : D_IN, D_H);

  recon_loss<<<(B_ROWS * D_IN) / 2048, 256, 0, stream>>>(xhat, x, lossacc);
  finalize_k<<<1, 1, 0, stream>>>(lossacc, l0cnt, loss, l0);
}